// GCN_73443940762210
// MI455X (gfx1250) — compile-verified
//
#include <hip/hip_runtime.h>
#include <hip/hip_bf16.h>

#define N_NODES  50000
#define N_EDGES  625000
#define HIDDEN   128
#define N_GRAPHS 256
#define N_CLASSES 10

typedef __attribute__((ext_vector_type(2))) float v2f;
typedef __attribute__((ext_vector_type(8))) float v8f;

// ---------------------------------------------------------------------------
// 1) degree with self-loops:  deg[i] = 1 + #{e : dst[e]==i}
// ---------------------------------------------------------------------------
__global__ void k_deg_init(float* __restrict__ deg) {
    int i = blockIdx.x * blockDim.x + threadIdx.x;
    if (i < N_NODES) deg[i] = 1.0f;              // self loop
}

__global__ void k_deg_edges(const int* __restrict__ dst, float* __restrict__ deg) {
    int e = blockIdx.x * blockDim.x + threadIdx.x;
    if (e < N_EDGES) atomicAdd(&deg[dst[e]], 1.0f);
}

// dis[i] = rsqrt(deg[i])   (deg >= 1 always, matches reference's where/max)
__global__ void k_dis(float* __restrict__ dis) {
    int i = blockIdx.x * blockDim.x + threadIdx.x;
    if (i < N_NODES) dis[i] = rsqrtf(dis[i]);
}

// ---------------------------------------------------------------------------
// 2) layer-1 aggregation is SCALAR (in_features == 1):
//    agg1[i] = x[i]*dis[i]^2 + sum_{e: dst=i} x[src]*dis[src]*dis[dst]
// ---------------------------------------------------------------------------
__global__ void k_agg1_init(const float* __restrict__ x, const float* __restrict__ dis,
                            float* __restrict__ agg1) {
    int i = blockIdx.x * blockDim.x + threadIdx.x;
    if (i < N_NODES) { float d = dis[i]; agg1[i] = x[i] * d * d; }
}

__global__ void k_agg1_edges(const int* __restrict__ src, const int* __restrict__ dst,
                             const float* __restrict__ x, const float* __restrict__ dis,
                             float* __restrict__ agg1) {
    int e = blockIdx.x * blockDim.x + threadIdx.x;
    if (e < N_EDGES) {
        int s = src[e], d = dst[e];
        atomicAdd(&agg1[d], x[s] * dis[s] * dis[d]);
    }
}

// ---------------------------------------------------------------------------
// 3) layer-2 aggregation of h1 (A @ h1), with h1 recomputed on the fly:
//    h1[i][f] = relu(agg1[i]*W1[f] + b1[f])
//    agg2[i][:] = h1[i][:]*dis[i]^2  +  sum_{e: dst=i} h1[src][:]*norm[e]
// ---------------------------------------------------------------------------
__global__ void k_agg2_init(const float* __restrict__ agg1, const float* __restrict__ dis,
                            const float* __restrict__ W1, const float* __restrict__ b1,
                            float* __restrict__ agg2) {
    int idx = blockIdx.x * blockDim.x + threadIdx.x;   // N_NODES*HIDDEN total
    int i = idx >> 7, f = idx & 127;
    float d  = dis[i];
    float h1 = fmaxf(agg1[i] * W1[f] + b1[f], 0.0f);
    agg2[idx] = h1 * d * d;
}

#define EDGES_PER_BLOCK 64
__global__ void k_agg2_edges(const int* __restrict__ src, const int* __restrict__ dst,
                             const float* __restrict__ agg1, const float* __restrict__ dis,
                             const float* __restrict__ W1, const float* __restrict__ b1,
                             float* __restrict__ agg2) {
    __shared__ float W1s[HIDDEN], b1s[HIDDEN];
    int tid  = threadIdx.x;            // 256 threads: 2 edge-slots x 128 features
    int f    = tid & 127;
    int slot = tid >> 7;
    if (tid < HIDDEN) { W1s[tid] = W1[tid]; b1s[tid] = b1[tid]; }
    __syncthreads();
    int ebase = blockIdx.x * EDGES_PER_BLOCK;
    #pragma unroll 4
    for (int it = 0; it < EDGES_PER_BLOCK / 2; ++it) {
        int e = ebase + it * 2 + slot;
        if (e < N_EDGES) {
            int s = src[e], d = dst[e];
            float nrm = dis[s] * dis[d];                      // norm[e]
            float h1  = fmaxf(agg1[s] * W1s[f] + b1s[f], 0.0f); // gather = 1 scalar
            atomicAdd(&agg2[(size_t)d * HIDDEN + f], h1 * nrm);
        }
    }
}

// ---------------------------------------------------------------------------
// 4) h2 = relu(agg2 @ W2 + b2)  via V_WMMA_F32_16X16X4_F32
//    grid: 3125 blocks (M-tiles of 16 rows), 8 waves/block (one 16-col N-tile each)
// ---------------------------------------------------------------------------
__global__ void __launch_bounds__(256)
k_gemm_wmma(const float* __restrict__ A, const float* __restrict__ W2,
            const float* __restrict__ b2, float* __restrict__ h2) {
    int wave = threadIdx.x >> 5;        // 0..7  -> N tile
    int lane = threadIdx.x & 31;
    int half = lane >> 4;               // 0: K={k0,k0+1}, 1: K={k0+2,k0+3}
    int l    = lane & 15;
    int m16  = blockIdx.x * 16;
    int mrow = m16 + l;                 // A-matrix row held by this lane
    int ncol = wave * 16 + l;           // B/C/D column held by this lane

    const float* Arow = A  + (size_t)mrow * HIDDEN + half * 2;       // A[mrow][k0+2h ..]
    const float* Bcol = W2 + (size_t)(half * 2) * HIDDEN + ncol;     // W2[k0+2h][ncol]

    v8f acc = {};
    #pragma unroll
    for (int k0 = 0; k0 < HIDDEN; k0 += 4) {
        v2f a, b;
        a.x = Arow[k0];
        a.y = Arow[k0 + 1];
        b.x = Bcol[(size_t)k0 * HIDDEN];
        b.y = Bcol[(size_t)(k0 + 1) * HIDDEN];
        acc = __builtin_amdgcn_wmma_f32_16x16x4_f32(
                  /*neg_a=*/false, a, /*neg_b=*/false, b,
                  /*c_mod=*/(short)0, acc, /*reuse_a=*/false, /*reuse_b=*/false);
    }

    float bias = b2[ncol];
    int M0 = m16 + half * 8;            // D layout: vgpr r -> rows r (lanes<16) / r+8
    #pragma unroll
    for (int r = 0; r < 8; ++r) {
        float v = acc[r] + bias;
        h2[(size_t)(M0 + r) * HIDDEN + ncol] = fmaxf(v, 0.0f);
    }
}

// ---------------------------------------------------------------------------
// 5) per-graph max/mean pooling + output GEMM + softmax (one block per graph)
//    batch_index is sorted -> binary-search segment boundaries
// ---------------------------------------------------------------------------
__device__ int lower_bound_batch(const int* __restrict__ batch, int key) {
    int lo = 0, hi = N_NODES;
    while (lo < hi) { int mid = (lo + hi) >> 1; if (batch[mid] < key) lo = mid + 1; else hi = mid; }
    return lo;
}

__global__ void __launch_bounds__(HIDDEN)
k_pool_head(const float* __restrict__ h2, const int* __restrict__ batch,
            const float* __restrict__ Wout, const float* __restrict__ bout,
            float* __restrict__ out) {
    int g = blockIdx.x;
    int f = threadIdx.x;                        // one thread per feature column
    __shared__ int   s_start, s_end;
    __shared__ float sm_max[HIDDEN], sm_mean[HIDDEN], logits[N_CLASSES];

    if (f == 0) {
        s_start = lower_bound_batch(batch, g);
        s_end   = lower_bound_batch(batch, g + 1);
    }
    __syncthreads();
    int start = s_start, end = s_end;

    float mx = -3.402823466e38f, sum = 0.0f;
    for (int i = start; i < end; ++i) {
        float v = h2[(size_t)i * HIDDEN + f];
        mx  = fmaxf(mx, v);
        sum += v;
    }
    float cnt = (float)((end - start) > 0 ? (end - start) : 1);
    sm_max[f]  = mx;
    sm_mean[f] = sum / cnt;
    __syncthreads();

    if (f < N_CLASSES) {
        float acc = bout[f];
        #pragma unroll 4
        for (int j = 0; j < HIDDEN; ++j) acc += sm_max[j]  * Wout[j * N_CLASSES + f];
        #pragma unroll 4
        for (int j = 0; j < HIDDEN; ++j) acc += sm_mean[j] * Wout[(HIDDEN + j) * N_CLASSES + f];
        logits[f] = acc;
    }
    __syncthreads();

    if (f == 0) {
        float m = logits[0];
        for (int c = 1; c < N_CLASSES; ++c) m = fmaxf(m, logits[c]);
        float e[N_CLASSES], s = 0.0f;
        for (int c = 0; c < N_CLASSES; ++c) { e[c] = expf(logits[c] - m); s += e[c]; }
        float inv = 1.0f / s;
        for (int c = 0; c < N_CLASSES; ++c) out[(size_t)g * N_CLASSES + c] = e[c] * inv;
    }
}

// ---------------------------------------------------------------------------
extern "C" void kernel_launch(void* const* d_in, const int* in_sizes, int n_in,
                              void* d_out, int out_size, void* d_ws, size_t ws_size,
                              hipStream_t stream) {
    const float* x     = (const float*)d_in[0];
    const int*   ei    = (const int*)  d_in[1];   // [2, E] row-major
    const int*   batch = (const int*)  d_in[2];
    const float* W1    = (const float*)d_in[3];
    const float* b1    = (const float*)d_in[4];
    const float* W2    = (const float*)d_in[5];
    const float* b2    = (const float*)d_in[6];
    const float* Wout  = (const float*)d_in[7];
    const float* bout  = (const float*)d_in[8];
    float*       out   = (float*)d_out;

    const int* src = ei;
    const int* dst = ei + N_EDGES;

    // workspace layout (floats)
    float* ws   = (float*)d_ws;
    float* agg2 = ws;                                        // N*H
    float* h2   = ws + (size_t)N_NODES * HIDDEN;             // N*H
    float* dis  = ws + (size_t)2 * N_NODES * HIDDEN;         // N (deg -> dis in place)
    float* agg1 = dis + N_NODES;                             // N

    const int TB = 256;
    const int gN = (N_NODES + TB - 1) / TB;                  // 196
    const int gE = (N_EDGES + TB - 1) / TB;                  // 2442

    k_deg_init  <<<gN, TB, 0, stream>>>(dis);
    k_deg_edges <<<gE, TB, 0, stream>>>(dst, dis);
    k_dis       <<<gN, TB, 0, stream>>>(dis);

    k_agg1_init <<<gN, TB, 0, stream>>>(x, dis, agg1);
    k_agg1_edges<<<gE, TB, 0, stream>>>(src, dst, x, dis, agg1);

    k_agg2_init <<<(N_NODES * HIDDEN) / TB, TB, 0, stream>>>(agg1, dis, W1, b1, agg2);
    k_agg2_edges<<<(N_EDGES + EDGES_PER_BLOCK - 1) / EDGES_PER_BLOCK, TB, 0, stream>>>(
        src, dst, agg1, dis, W1, b1, agg2);

    k_gemm_wmma <<<N_NODES / 16, TB, 0, stream>>>(agg2, W2, b2, h2);  // 3125 blocks

    k_pool_head <<<N_GRAPHS, HIDDEN, 0, stream>>>(h2, batch, Wout, bout, out);
}